// EqPBCNN_15410342658739
// MI455X (gfx1250) — compile-verified
//
#include <hip/hip_runtime.h>
#include <hip/hip_bf16.h>

// ---------------------------------------------------------------------------
// EqPBCNN forward for MI455X (gfx1250).
//
//  * 175 (m,n) index pairs -> per batch elt a 700-wide real feature vector
//    (4 features per pair: Fr_p0, Fi_p0, Fr_p1, Fi_p1). Layer 1 is a
//    65536 x 700 x 8 real GEMM done with V_WMMA_F32_16X16X4_F32: one WMMA
//    per pair (K=4), A operand computed in registers (lanes 0-15 = pol0
//    features / rows 0-15, lanes 16-31 = pol1 features -> matches the ISA
//    16x4 f32 A layout exactly). Pair table padded to 176 (even) with a
//    zero-weight dummy so the loop unrolls 2x onto two independent
//    accumulators (no WMMA->WMMA RAW chain, no accumulator register moves).
//  * Packed +/-(W1r,W1i) B matrix built once per block in LDS, zero-padded
//    to all 16 columns so the B ds_load_b64 is unconditional (no per-iter
//    EXEC save/restore) and bank-conflict-free.
//  * x tile staged in LDS so the En/Emn gathers are single b128 LDS loads.
//  * Layers 2/3 (complex 2->10->1) + leaky ReLU + output done per lane
//    (lane = (batch-in-tile, pol)).
// ---------------------------------------------------------------------------

typedef __attribute__((ext_vector_type(2))) float v2f;
typedef __attribute__((ext_vector_type(8))) float v8f;

#define LHALF   20
#define MMODES  41
#define NMODES  2
#define H1DIM   2
#define H2DIM   10
#define BATCHSZ 65536

#define WAVES_PER_BLOCK 4
#define TILE            16
#define XF              (MMODES * 4)   // floats per staged batch element
#define H1STRIDE        20             // padded row stride for h1 spill tile

struct PairTab {
    unsigned pack[442];
    int cnt;
};

// Replicates the Python index-set construction exactly:
//   m in [-L,L], n in [-L,L], n>=m, |m*n|<=L, |m+n|<=L   (RHO*L == L here)
constexpr PairTab buildTab() {
    PairTab t{};
    int c = 0;
    for (int m = -LHALF; m <= LHALF; ++m) {
        for (int n = -LHALF; n <= LHALF; ++n) {
            if (n < m) continue;
            long mn = (long)m * (long)n; if (mn < 0) mn = -mn;
            int s = m + n; if (s < 0) s = -s;
            if (mn <= LHALF && s <= LHALF) {
                unsigned en  = (unsigned)(LHALF + n);
                unsigned emn = (unsigned)(LHALF + m + n);
                unsigned em  = (unsigned)(LHALF + m);
                unsigned sy  = (m != n) ? 1u : 0u;
                t.pack[c] = en | (emn << 6) | (em << 12) | (sy << 18);
                ++c;
            }
        }
    }
    t.cnt = c;
    t.pack[c] = t.pack[0];   // dummy pad entry (its B columns are zero)
    return t;
}

constexpr PairTab TAB = buildTab();
constexpr int HDIM  = TAB.cnt;                    // = 175
constexpr int HDIMP = HDIM + (HDIM & 1);          // = 176 (even, for 2x unroll)
static_assert(HDIM > 0 && HDIM <= 441, "HDIM bounds");

__device__ __constant__ PairTab dTAB = TAB;

__device__ __forceinline__ float lrelu(float x) {
    return x > 0.f ? x : 0.01f * x;
}

__device__ __forceinline__ v8f wmma_step(const float* __restrict__ xe,
                                         const float* __restrict__ bp,
                                         int p, unsigned pk, v8f acc)
{
    int iEn  =  pk        & 63;
    int iEmn = (pk >> 6)  & 63;
    int iEm  = (pk >> 12) & 63;
    float sym = (pk & (1u << 18)) ? 2.f : 1.f;

    const float4 En  = *(const float4*)(xe + iEn  * 4);  // (re0,im0,re1,im1)
    const float4 Emn = *(const float4*)(xe + iEmn * 4);
    const v2f    Em  = *(const v2f*)  (xe + iEm * 4 + 2 * p);

    // A' = sum_p En_p * conj(Emn_p)   (same for both pols after mode-flip add)
    float Ar = En.x * Emn.x + En.y * Emn.y + En.z * Emn.z + En.w * Emn.w;
    float Ai = En.y * Emn.x - En.x * Emn.y + En.w * Emn.z - En.z * Emn.w;
    // F_p = A' * Em_p * sym
    v2f a;
    a.x = (Ar * Em.x - Ai * Em.y) * sym;
    a.y = (Ar * Em.y + Ai * Em.x) * sym;

    // B operand: lane holds column c (=lane&15), K-half (=lane>>4);
    // cols 8..15 are zero-filled in LDS -> unconditional load.
    v2f b = *(const v2f*)bp;

    return __builtin_amdgcn_wmma_f32_16x16x4_f32(
        false, a, false, b, (short)0, acc, false, false);
}

__global__ __launch_bounds__(WAVES_PER_BLOCK * 32)
void eqpbcnn_kernel(const float* __restrict__ x_real,
                    const float* __restrict__ x_imag,
                    const float* __restrict__ task_info,
                    const float* __restrict__ W1r, const float* __restrict__ W1i,
                    const float* __restrict__ W2r, const float* __restrict__ W2i,
                    const float* __restrict__ W3r, const float* __restrict__ W3i,
                    float* __restrict__ out)
{
    extern __shared__ float lds[];
    float* Blds = lds;                                   // HDIMP*16*4 floats
    float* xall = lds + HDIMP * 64;

    const int tid  = threadIdx.x;
    const int lane = tid & 31;
    const int wave = tid >> 5;

    float* xw  = xall + wave * (TILE * XF);
    float* h1w = xall + WAVES_PER_BLOCK * (TILE * XF) + wave * (TILE * H1STRIDE);

    // ---- build packed layer-1 B matrix (K=4*HDIMP rows x 16 cols) in LDS ----
    // storage: Blds[j*64 + c*4 + t], c = 4*p + 2*o + part (c>=8 zero),
    //          t = 2*pf + partf
    for (int idx = tid; idx < HDIMP * 64; idx += blockDim.x) {
        int t = idx & 3;
        int c = (idx >> 2) & 15;
        int j = idx >> 6;
        float v = 0.f;
        if (c < 8 && j < HDIM) {
            int p = c >> 2, o = (c >> 1) & 1, part = c & 1;
            int pf = t >> 1, partf = t & 1;
            if (pf == p) {
                float wr = W1r[(p * H1DIM + o) * HDIM + j];
                float wi = W1i[(p * H1DIM + o) * HDIM + j];
                v = (part == 0) ? (partf == 0 ? wr : -wi)
                                : (partf == 0 ? wi :  wr);
            }
        }
        Blds[idx] = v;
    }
    __syncthreads();

    const int tile = blockIdx.x * WAVES_PER_BLOCK + wave;
    const int base = tile * TILE;

    // ---- stage x tile: xw[elt][mode][pol][re/im] ----
    const float* xr = x_real + (size_t)base * (MMODES * NMODES);
    const float* xi = x_imag + (size_t)base * (MMODES * NMODES);
    for (int idx = lane; idx < TILE * MMODES * NMODES; idx += 32) {
        float a = xr[idx];
        float b = xi[idx];
        int e = idx / (MMODES * NMODES);
        int t = idx - e * (MMODES * NMODES);
        float* d = &xw[e * XF + t * 2];
        d[0] = a;
        d[1] = b;
    }
    __builtin_amdgcn_wave_barrier();
    asm volatile("s_wait_dscnt 0x0" ::: "memory");

    const int r = lane & 15;    // batch row in tile (A rows / B column)
    const int p = lane >> 4;    // pol for A features; K-half for B operand
    const float* xe = xw + r * XF;
    const float* bp = Blds + r * 4 + p * 2;   // + j*64 per pair

    v8f acc0 = {};
    v8f acc1 = {};
    for (int j = 0; j < HDIMP; j += 2) {
        unsigned pk0 = dTAB.pack[j];
        unsigned pk1 = dTAB.pack[j + 1];
        acc0 = wmma_step(xe, bp + (size_t)j * 64,       p, pk0, acc0);
        acc1 = wmma_step(xe, bp + (size_t)(j + 1) * 64, p, pk1, acc1);
    }
    v8f acc = acc0 + acc1;

    // ---- spill C (16x16) to padded LDS tile to re-map rows->lanes ----
#pragma unroll
    for (int v = 0; v < 8; ++v) {
        int row = p * 8 + v;            // lanes 0-15: rows 0-7; 16-31: rows 8-15
        h1w[row * H1STRIDE + r] = acc[v];
    }
    __builtin_amdgcn_wave_barrier();
    asm volatile("s_wait_dscnt 0x0" ::: "memory");

    // ---- per-lane tail: lane = (batch row r, pol p) ----
    float4 h = *(const float4*)(h1w + r * H1STRIDE + 4 * p);
    float a0 = lrelu(h.x), b0 = lrelu(h.y);   // h1[o=0] re/im
    float a1 = lrelu(h.z), b1 = lrelu(h.w);   // h1[o=1] re/im

    const float* w2r = W2r + p * (H2DIM * H1DIM);
    const float* w2i = W2i + p * (H2DIM * H1DIM);
    const float* w3r = W3r + p * H2DIM;
    const float* w3i = W3i + p * H2DIM;

    float Er = 0.f, Ei = 0.f;
#pragma unroll
    for (int q = 0; q < H2DIM; ++q) {
        float c0 = w2r[q * 2 + 0], c1 = w2r[q * 2 + 1];
        float d0 = w2i[q * 2 + 0], d1 = w2i[q * 2 + 1];
        float hr = a0 * c0 - b0 * d0 + a1 * c1 - b1 * d1;
        float hi = a0 * d0 + b0 * c0 + a1 * d1 + b1 * c1;
        hr = lrelu(hr);
        hi = lrelu(hi);
        float wr = w3r[q], wi = w3i[q];
        Er += hr * wr - hi * wi;
        Ei += hr * wi + hi * wr;
    }

    int bidx = base + r;
    float P = exp10f(task_info[bidx * 4] * 0.1f) * 0.5f;
    float xLr = xw[r * XF + (LHALF * 2 + p) * 2 + 0];
    float xLi = xw[r * XF + (LHALF * 2 + p) * 2 + 1];

    float2 res;
    res.x = xLr + Er * P;
    res.y = xLi + Ei * P;
    *(float2*)(out + ((size_t)bidx * NMODES + p) * 2) = res;
}

extern "C" void kernel_launch(void* const* d_in, const int* in_sizes, int n_in,
                              void* d_out, int out_size, void* d_ws, size_t ws_size,
                              hipStream_t stream) {
    const float* x_real    = (const float*)d_in[0];
    const float* x_imag    = (const float*)d_in[1];
    const float* task_info = (const float*)d_in[2];
    const float* W1r       = (const float*)d_in[3];
    const float* W1i       = (const float*)d_in[4];
    const float* W2r       = (const float*)d_in[5];
    const float* W2i       = (const float*)d_in[6];
    const float* W3r       = (const float*)d_in[7];
    const float* W3i       = (const float*)d_in[8];
    float* out             = (float*)d_out;

    const int blocks = BATCHSZ / (TILE * WAVES_PER_BLOCK);   // 1024
    const size_t ldsFloats = (size_t)HDIMP * 64
                           + (size_t)WAVES_PER_BLOCK * TILE * XF
                           + (size_t)WAVES_PER_BLOCK * TILE * H1STRIDE;
    const size_t ldsBytes = ldsFloats * sizeof(float);       // ~90 KB

    eqpbcnn_kernel<<<blocks, WAVES_PER_BLOCK * 32, ldsBytes, stream>>>(
        x_real, x_imag, task_info, W1r, W1i, W2r, W2i, W3r, W3i, out);
}